// EGraphMAE_71468255805603
// MI455X (gfx1250) — compile-verified
//
#include <hip/hip_runtime.h>
#include <math.h>

// ---------------------------------------------------------------------------
// EGraphMAE forward for MI455X (gfx1250, wave32).
// Dense GEMMs: V_WMMA_F32_16X16X4_F32, one wave per 16x(NT*16) output strip
// (NT accumulator tiles -> NT-fold A-fragment reuse, NT compile-time so the
// accumulators never spill), A tile staged into LDS via gfx1250 async
// global->LDS copies (ASYNCcnt).  Edge softmax/aggregation uses native f32
// global atomics.
// Input pointer order = setup_inputs() dict insertion order, params flattened
// depth-first in insertion order:
//   0:x 1:e 2:mask_token 3:W_e2d 4:W_e2d_e
//   5..16  enc[0]: Wn Wni Wnj Wf b attn Wres Wres_e g_n b_n g_e b_e
//   17..28 enc[1]: (same order)
//   29..40 dec[0]: (same order)
//   41:src 42:dst 43:keep_idx 44:mask_nodes 45:token_nodes 46:noise_nodes
//   47:noise_src 48:n_nodes
// ---------------------------------------------------------------------------

typedef __attribute__((ext_vector_type(2))) float v2f;
typedef __attribute__((ext_vector_type(8))) float v8f;

#if defined(__gfx1250__) && \
    __has_builtin(__builtin_amdgcn_global_load_async_to_lds_b128) && \
    __has_builtin(__builtin_amdgcn_s_wait_asynccnt)
#define USE_ASYNC_LDS 1
typedef int v4i __attribute__((vector_size(16)));       // 4 x i32, matches builtin proto
typedef __attribute__((address_space(1))) v4i gv4i;     // global-AS vector
typedef __attribute__((address_space(3))) v4i lv4i;     // LDS-AS vector
#endif

#define NN      20000
#define INDIM   128
#define NHID    256
#define EFD     16
#define EHD     64
#define EKEEP   320000
#define EU      340000     // EKEEP + NN self loops
#define NMASK   10000
#define NTOKEN  8500
#define NNOISE  1500

// ------------------------------------------------------------------ helpers
__global__ void k_fill(float* __restrict__ p, float v, int n) {
  int i = blockIdx.x * blockDim.x + threadIdx.x;
  if (i < n) p[i] = v;
}

__global__ void k_token(const int* __restrict__ tok, const float* __restrict__ mask_token,
                        float* __restrict__ ux) {
  int i = blockIdx.x * blockDim.x + threadIdx.x;
  if (i >= NTOKEN * INDIM) return;
  int t = i / INDIM, d = i % INDIM;
  ux[(size_t)tok[t] * INDIM + d] = mask_token[d];   // set(0) then add(token) == token
}

__global__ void k_noise(const int* __restrict__ nodes, const int* __restrict__ nsrc,
                        const float* __restrict__ x, float* __restrict__ ux) {
  int i = blockIdx.x * blockDim.x + threadIdx.x;
  if (i >= NNOISE * INDIM) return;
  int t = i / INDIM, d = i % INDIM;
  ux[(size_t)nodes[t] * INDIM + d] = x[(size_t)nsrc[t] * INDIM + d];
}

__global__ void k_ue(const float* __restrict__ e, const int* __restrict__ keep,
                     float* __restrict__ ue) {
  int i = blockIdx.x * blockDim.x + threadIdx.x;
  if (i >= EU * EFD) return;
  int ei = i / EFD, d = i % EFD;
  ue[i] = (ei < EKEEP) ? e[(size_t)keep[ei] * EFD + d] : 0.0f;
}

__global__ void k_zero_rows(const int* __restrict__ rows, float* __restrict__ p, int D, int n) {
  int i = blockIdx.x * blockDim.x + threadIdx.x;
  if (i >= n) return;
  p[(size_t)rows[i / D] * D + (i % D)] = 0.0f;
}

__device__ inline void atomicMaxF(float* addr, float val) {
  int* ia = (int*)addr;
  int old = *ia;
  while (__int_as_float(old) < val) {
    int assumed = old;
    old = atomicCAS(ia, assumed, __float_as_int(val));
    if (old == assumed) break;
  }
}

// ------------------------------------------------------- FP32 WMMA GEMM
// C[M x Nw] = A[M x K] @ W[K x Nw]; M,K,Nw multiples of 16 (true for every
// GEMM in this model).  One wave per 16(M) x NT*16(N) strip; NT is a template
// parameter so the NT accumulator tiles fully unroll with no branches and
// stay in VGPRs.  A tile (16x16) staged in LDS via async global->LDS b128.
// A operand layout (ISA 7.12.2, 32-bit 16x4): lanes 0-15 hold K=k0,k0+1 and
// lanes 16-31 hold K=k0+2,k0+3 for row M=lane%16.  B mirrors with N across
// lanes.  C/D: VGPR r holds M=r (lanes 0-15) / M=r+8 (lanes 16-31), N=lane%16.
template <int NT>
__global__ void k_gemm_wmma(const float* __restrict__ A, const float* __restrict__ W,
                            float* __restrict__ C, int M, int K, int Nw) {
  __shared__ float As[256];                 // 16 x 16 A tile
  const int lane = threadIdx.x;             // wave32
  const int half = lane >> 4;               // 0: K+0/1  1: K+2/3
  const int l16  = lane & 15;
  const int bm = blockIdx.y;
  const int n0 = blockIdx.x * (NT * 16);    // first column of this strip
  v8f acc[NT] = {};

  for (int k0 = 0; k0 < K; k0 += 16) {
#if defined(USE_ASYNC_LDS)
    // 16x16 f32 tile = 64 x b128; 2 async copies per lane (ASYNCcnt domain).
#pragma unroll
    for (int i = 0; i < 2; ++i) {
      int q  = lane + 32 * i;               // b128 element id 0..63
      int r  = q >> 2;                      // row 0..15
      int c4 = (q & 3) << 2;                // col 0,4,8,12
      __builtin_amdgcn_global_load_async_to_lds_b128(
          (gv4i*)(A + (size_t)(bm * 16 + r) * K + (k0 + c4)),
          (lv4i*)(As + r * 16 + c4), 0, 0);
    }
    __builtin_amdgcn_s_wait_asynccnt(0);
    __syncthreads();
#else
#pragma unroll
    for (int i = 0; i < 8; ++i) {           // cooperative 16x16 load, coalesced
      int t = lane + 32 * i;
      As[t] = A[(size_t)(bm * 16 + (t >> 4)) * K + (k0 + (t & 15))];
    }
    __syncthreads();
#endif
    if (k0 + 16 < K)                        // prefetch next K-chunk of W panel
      __builtin_prefetch(W + (size_t)(k0 + 16) * Nw + n0 + lane, 0, 1);

#pragma unroll
    for (int kk = 0; kk < 16; kk += 4) {
      v2f a;
      a.x = As[l16 * 16 + kk + 2 * half];
      a.y = As[l16 * 16 + kk + 2 * half + 1];
      const float* wrow = W + (size_t)(k0 + kk + 2 * half) * Nw + n0 + l16;
#pragma unroll
      for (int nt = 0; nt < NT; ++nt) {
        v2f b;
        b.x = wrow[nt * 16];
        b.y = wrow[nt * 16 + Nw];
        acc[nt] = __builtin_amdgcn_wmma_f32_16x16x4_f32(false, a, false, b,
                                                        (short)0, acc[nt],
                                                        false, false);
      }
    }
    __syncthreads();
  }

  const size_t row0 = (size_t)bm * 16 + 8 * half;
#pragma unroll
  for (int nt = 0; nt < NT; ++nt) {
    const size_t col = (size_t)n0 + nt * 16 + l16;
#pragma unroll
    for (int r = 0; r < 8; ++r)
      C[(row0 + r) * Nw + col] = acc[nt][r];
  }
}

// ------------------------------------------------------------- edge kernels
// fij = lrelu( hni[src] + hnj[dst] + fW + b , 0.2 )
__global__ void k_edge_fij(const float* __restrict__ hni, const float* __restrict__ hnj,
                           const float* __restrict__ fW, const float* __restrict__ bias,
                           const int* __restrict__ src, const int* __restrict__ dst,
                           float* __restrict__ fij, int De, int n) {
  int i = blockIdx.x * blockDim.x + threadIdx.x;
  if (i >= n) return;
  int e = i / De, j = i % De;
  float v = hni[(size_t)src[e] * De + j] + hnj[(size_t)dst[e] * De + j] + fW[i] + bias[j];
  fij[i] = (v > 0.0f) ? v : 0.2f * v;
}

// logits[e,h] = sum_d fij[e,h,d]*attn[h,d];  mx[dst,h] = segment max
__global__ void k_edge_logits_max(const float* __restrict__ fij, const float* __restrict__ attn,
                                  const int* __restrict__ dst, float* __restrict__ lg,
                                  float* __restrict__ mx, int H, int oe, int n) {
  int i = blockIdx.x * blockDim.x + threadIdx.x;
  if (i >= n) return;
  int e = i / H, h = i % H;
  const float* fp = fij + (size_t)e * (H * oe) + h * oe;
  const float* ap = attn + h * oe;
  float s = 0.0f;
  for (int d = 0; d < oe; ++d) s += fp[d] * ap[d];
  lg[i] = s;
  atomicMaxF(&mx[(size_t)dst[e] * H + h], s);
}

// w = exp(logit - mx[dst]);  den[dst,h] += w   (w overwrites logits)
__global__ void k_edge_softmax(float* __restrict__ lg, const float* __restrict__ mx,
                               float* __restrict__ den, const int* __restrict__ dst,
                               int H, int n) {
  int i = blockIdx.x * blockDim.x + threadIdx.x;
  if (i >= n) return;
  int e = i / H, h = i % H;
  float m = mx[(size_t)dst[e] * H + h];
  if (!(m > -INFINITY && m < INFINITY)) m = 0.0f;   // isfinite guard (ref semantics)
  float wv = expf(lg[i] - m);
  lg[i] = wv;
  atomicAdd(&den[(size_t)dst[e] * H + h], wv);
}

// hout[dst,h,d] += hw[src,h,d] * w[e,h]/(den[dst,h]+1e-9)
__global__ void k_edge_aggregate(const float* __restrict__ wgt, const float* __restrict__ den,
                                 const float* __restrict__ hw, const int* __restrict__ src,
                                 const int* __restrict__ dst, float* __restrict__ hout,
                                 int H, int on, int n) {
  int i = blockIdx.x * blockDim.x + threadIdx.x;
  if (i >= n) return;
  int Dn = H * on;
  int e = i / Dn, r = i % Dn, h = r / on;
  int de = dst[e], se = src[e];
  float alpha = wgt[(size_t)e * H + h] / (den[(size_t)de * H + h] + 1e-9f);
  atomicAdd(&hout[(size_t)de * Dn + r], hw[(size_t)se * Dn + r] * alpha);
}

__global__ void k_add(const float* __restrict__ a, float* __restrict__ b, int n) {
  int i = blockIdx.x * blockDim.x + threadIdx.x;
  if (i < n) b[i] += a[i];
}

// ------------------------------------------------- layernorm (+ ELU), 1 wave/row
__global__ void k_ln(float* __restrict__ p, const float* __restrict__ g,
                     const float* __restrict__ b, int D, int act) {
  const size_t row = blockIdx.x;
  const int lane = threadIdx.x;
  float s = 0.0f, s2 = 0.0f;
  for (int j = lane; j < D; j += 32) {
    float v = p[row * D + j];
    s += v; s2 += v * v;
  }
  for (int off = 16; off > 0; off >>= 1) {
    s  += __shfl_xor(s, off, 32);
    s2 += __shfl_xor(s2, off, 32);
  }
  float m   = s / D;
  float var = s2 / D - m * m;
  float inv = rsqrtf(var + 1e-5f);
  for (int j = lane; j < D; j += 32) {
    float v = (p[row * D + j] - m) * inv * g[j] + b[j];
    if (act) v = (v > 0.0f) ? v : (expf(v) - 1.0f);   // ELU
    p[row * D + j] = v;
  }
}

// ---------------------------------------------------------------- final loss
__global__ void k_loss(const float* __restrict__ x, const float* __restrict__ hd,
                       const int* __restrict__ mask_nodes, float* __restrict__ out) {
  const int m = mask_nodes[blockIdx.x];
  const int lane = threadIdx.x;
  float dot = 0.0f, nx = 0.0f, nr = 0.0f;
  for (int j = lane; j < INDIM; j += 32) {
    float a = x[(size_t)m * INDIM + j];
    float b = hd[(size_t)m * INDIM + j];
    dot += a * b; nx += a * a; nr += b * b;
  }
  for (int off = 16; off > 0; off >>= 1) {
    dot += __shfl_xor(dot, off, 32);
    nx  += __shfl_xor(nx,  off, 32);
    nr  += __shfl_xor(nr,  off, 32);
  }
  if (lane == 0) {
    float dn = fmaxf(sqrtf(nx), 1e-12f) * fmaxf(sqrtf(nr), 1e-12f);
    float c = 1.0f - dot / dn;
    atomicAdd(out, c * c * c * (1.0f / NMASK));       // ALPHA = 3
  }
}

// ------------------------------------------------------------------- host
struct LayerW {
  const float *Wn, *Wni, *Wnj, *Wf, *b, *attn, *Wres, *Wres_e, *g_n, *b_n, *g_e, *b_e;
};

static inline LayerW layer_from(void* const* d_in, int base) {
  LayerW L;
  L.Wn     = (const float*)d_in[base + 0];
  L.Wni    = (const float*)d_in[base + 1];
  L.Wnj    = (const float*)d_in[base + 2];
  L.Wf     = (const float*)d_in[base + 3];
  L.b      = (const float*)d_in[base + 4];
  L.attn   = (const float*)d_in[base + 5];
  L.Wres   = (const float*)d_in[base + 6];
  L.Wres_e = (const float*)d_in[base + 7];
  L.g_n    = (const float*)d_in[base + 8];
  L.b_n    = (const float*)d_in[base + 9];
  L.g_e    = (const float*)d_in[base + 10];
  L.b_e    = (const float*)d_in[base + 11];
  return L;
}

static inline void gemm(hipStream_t s, const float* A, const float* W, float* C,
                        int M, int K, int Nw) {
  if ((Nw & 63) == 0) {                       // 64, 128, 256 wide panels
    dim3 g(Nw / 64, M / 16);
    k_gemm_wmma<4><<<g, 32, 0, s>>>(A, W, C, M, K, Nw);
  } else {                                    // 16-wide panels (decoder edges)
    dim3 g(Nw / 16, M / 16);
    k_gemm_wmma<1><<<g, 32, 0, s>>>(A, W, C, M, K, Nw);
  }
}

static void egat_layer(hipStream_t s, const LayerW& L,
                       const float* h, const float* f, float* h_next, float* f_next,
                       float* hni, float* hnj, float* hw, float* fW, float* fij,
                       float* lg, float* mx, float* den,
                       const int* src, const int* dst,
                       int in_n, int in_e, int out_n, int out_e, int H,
                       bool act, bool norm) {
  const int Dn = H * out_n, De = H * out_e;
  gemm(s, h, L.Wni,    hni,    NN, in_n, De);
  gemm(s, h, L.Wnj,    hnj,    NN, in_n, De);
  gemm(s, h, L.Wn,     hw,     NN, in_n, Dn);
  gemm(s, h, L.Wres,   h_next, NN, in_n, Dn);   // residual = hout init
  gemm(s, f, L.Wf,     fW,     EU, in_e, De);
  gemm(s, f, L.Wres_e, f_next, EU, in_e, De);   // edge residual = eout init

  const int nfe = EU * De;
  k_edge_fij<<<(nfe + 255) / 256, 256, 0, s>>>(hni, hnj, fW, L.b, src, dst, fij, De, nfe);

  const int nh = NN * H;
  k_fill<<<(nh + 255) / 256, 256, 0, s>>>(mx, -INFINITY, nh);
  k_fill<<<(nh + 255) / 256, 256, 0, s>>>(den, 0.0f, nh);

  const int neh = EU * H;
  k_edge_logits_max<<<(neh + 255) / 256, 256, 0, s>>>(fij, L.attn, dst, lg, mx, H, out_e, neh);
  k_edge_softmax<<<(neh + 255) / 256, 256, 0, s>>>(lg, mx, den, dst, H, neh);

  const int nagg = EU * Dn;
  k_edge_aggregate<<<(nagg + 255) / 256, 256, 0, s>>>(lg, den, hw, src, dst, h_next, H, out_n, nagg);
  k_add<<<(nfe + 255) / 256, 256, 0, s>>>(fij, f_next, nfe);   // eout = fij + residual

  if (norm) {
    k_ln<<<NN, 32, 0, s>>>(h_next, L.g_n, L.b_n, Dn, act ? 1 : 0);
    k_ln<<<EU, 32, 0, s>>>(f_next, L.g_e, L.b_e, De, act ? 1 : 0);
  }
}

extern "C" void kernel_launch(void* const* d_in, const int* in_sizes, int n_in,
                              void* d_out, int out_size, void* d_ws, size_t ws_size,
                              hipStream_t stream) {
  (void)in_sizes; (void)n_in; (void)out_size; (void)ws_size;
  const float* x          = (const float*)d_in[0];
  const float* e          = (const float*)d_in[1];
  const float* mask_token = (const float*)d_in[2];
  const float* W_e2d      = (const float*)d_in[3];
  const float* W_e2d_e    = (const float*)d_in[4];
  LayerW enc0 = layer_from(d_in, 5);
  LayerW enc1 = layer_from(d_in, 17);
  LayerW dec0 = layer_from(d_in, 29);
  const int* src         = (const int*)d_in[41];
  const int* dst         = (const int*)d_in[42];
  const int* keep_idx    = (const int*)d_in[43];
  const int* mask_nodes  = (const int*)d_in[44];
  const int* token_nodes = (const int*)d_in[45];
  const int* noise_nodes = (const int*)d_in[46];
  const int* noise_src   = (const int*)d_in[47];

  // Workspace bump allocator (~427 MB total, f32).
  float* w = (float*)d_ws;
  size_t off = 0;
  auto alloc = [&](size_t n) { float* p = w + off; off += (n + 63) & ~(size_t)63; return p; };
  float* hA  = alloc((size_t)NN * 256);   // node ping
  float* hB  = alloc((size_t)NN * 256);   // node pong
  float* hni = alloc((size_t)NN * 64);
  float* hnj = alloc((size_t)NN * 64);
  float* hw  = alloc((size_t)NN * 256);
  float* fA  = alloc((size_t)EU * 64);    // edge ping
  float* fB  = alloc((size_t)EU * 64);    // edge pong
  float* fW  = alloc((size_t)EU * 64);
  float* fij = alloc((size_t)EU * 64);
  float* lg  = alloc((size_t)EU * 4);     // logits, then softmax weights
  float* mx  = alloc((size_t)NN * 4);
  float* den = alloc((size_t)NN * 4);

  // ---- masked input ux (into hA, N x 128) and ue (into fA, Eu x 16) ----
  (void)hipMemcpyAsync(hA, x, sizeof(float) * (size_t)NN * INDIM,
                       hipMemcpyDeviceToDevice, stream);
  k_token<<<(NTOKEN * INDIM + 255) / 256, 256, 0, stream>>>(token_nodes, mask_token, hA);
  k_noise<<<(NNOISE * INDIM + 255) / 256, 256, 0, stream>>>(noise_nodes, noise_src, x, hA);
  k_ue<<<(EU * EFD + 255) / 256, 256, 0, stream>>>(e, keep_idx, fA);

  // ---- encoder ----
  egat_layer(stream, enc0, hA, fA, hB, fB, hni, hnj, hw, fW, fij, lg, mx, den,
             src, dst, INDIM, EFD, 64, 16, 4, true, true);
  egat_layer(stream, enc1, hB, fB, hA, fA, hni, hnj, hw, fW, fij, lg, mx, den,
             src, dst, NHID, EHD, 64, 16, 4, true, true);

  // ---- encoder-to-decoder projection, mask rep ----
  gemm(stream, hA, W_e2d,   hB, NN, NHID, NHID);   // rep
  gemm(stream, fA, W_e2d_e, fB, EU, EHD,  EHD);    // rep_e
  k_zero_rows<<<(NMASK * NHID + 255) / 256, 256, 0, stream>>>(mask_nodes, hB, NHID, NMASK * NHID);

  // ---- decoder (H=1, no act/norm); node output N x 128 into hA ----
  egat_layer(stream, dec0, hB, fB, hA, fA, hni, hnj, hw, fW, fij, lg, mx, den,
             src, dst, NHID, EHD, INDIM, EFD, 1, false, false);

  // ---- scaled cosine error over masked nodes ----
  k_fill<<<1, 32, 0, stream>>>((float*)d_out, 0.0f, 1);
  k_loss<<<NMASK, 32, 0, stream>>>(x, hA, mask_nodes, (float*)d_out);
}